// Autoformer_45363444580643
// MI455X (gfx1250) — compile-verified
//
#include <hip/hip_runtime.h>

// ---------------------------------------------------------------------------
// Autoformer forward for MI455X (gfx1250, wave32).
// All dense contractions go through V_WMMA_F32_16X16X4_F32 (fp32 WMMA).
// GEMM waves compute a 32x64 strip (2 M-tiles x 4 N-tiles => 8 accumulators),
// with software-pipelined double-buffered operand loads so WMMAs overlap the
// L2 latency of the next K-group instead of draining loadcnt to zero.
// ---------------------------------------------------------------------------

typedef float v2f __attribute__((ext_vector_type(2)));
typedef float v8f __attribute__((ext_vector_type(8)));

// Model constants (match reference)
#define BN   16
#define LS   96      // SEQ_LEN
#define LBL  48      // LABEL_LEN
#define PRED 720     // PRED_LEN
#define LD   768     // LABEL_LEN + PRED_LEN
#define NS   321     // N_SERIES
#define DM   256     // D_MODEL
#define DF   1024    // D_FF
#define TOPK_ENC 13  // int(3*ln(96))
#define TOPK_DEC 19  // int(3*ln(768))
#define TK_STRIDE 32
#define MT   2       // M-tiles (16 rows each) per wave
#define NT   4       // N-tiles (16 cols each) per wave

// ---------------------------------------------------------------------------
// GEMM: C[z] = A[z](MxK) * B[z](NxK)^T (+bias[N]) (+C)  -- fp32 WMMA 16x16x4
// Requirements honored by all call sites: K multiple of 8. Row/col raggedness
// handled by clamped (non-divergent) loads + guarded stores.
// ---------------------------------------------------------------------------
__global__ void __launch_bounds__(32) gemm_xwt_wmma(
    const float* __restrict__ A, long sA,
    const float* __restrict__ Bw, long sB,
    float* C, long sC,
    const float* __restrict__ bias,
    int M, int N, int K, int accum)
{
    const int z = blockIdx.z;
    const float* Ab = A + (long)z * sA;
    const float* Bb = Bw + (long)z * sB;
    float* Cb = C + (long)z * sC;

    const int m0 = blockIdx.y << 5;        // 32 rows per wave
    const int n0 = blockIdx.x << 6;        // 64 cols per wave
    const int lane = threadIdx.x & 31;
    const int half = lane >> 4;            // 0 -> K pair {0,1}, 1 -> {2,3}
    const int l15  = lane & 15;
    const int kb   = half << 1;

    const float* ap[MT];
    #pragma unroll
    for (int m = 0; m < MT; ++m) {
        int ar = m0 + (m << 4) + l15;      // A row (clamped; stores guarded)
        if (ar >= M) ar = M - 1;
        ap[m] = Ab + (long)ar * K + kb;
    }
    const float* bp[NT];
    #pragma unroll
    for (int t = 0; t < NT; ++t) {
        int br = n0 + (t << 4) + l15;      // W row == C column (clamped)
        if (br >= N) br = N - 1;
        bp[t] = Bb + (long)br * K + kb;
    }

    v8f acc[MT][NT];
    #pragma unroll
    for (int m = 0; m < MT; ++m)
        #pragma unroll
        for (int t = 0; t < NT; ++t)
            acc[m][t] = (v8f){0.f,0.f,0.f,0.f,0.f,0.f,0.f,0.f};

    // Software pipeline: two K-groups (4 each) in flight.
    v2f a0[MT], a1[MT], b0[NT], b1[NT];
    #pragma unroll
    for (int m = 0; m < MT; ++m) { a0[m] = *(const v2f*)(ap[m]);
                                   a1[m] = *(const v2f*)(ap[m] + 4); }
    #pragma unroll
    for (int t = 0; t < NT; ++t) { b0[t] = *(const v2f*)(bp[t]);
                                   b1[t] = *(const v2f*)(bp[t] + 4); }

    for (int k = 8; k < K; k += 8) {
        // consume group 0 (loads issued one iteration ago), refill group 0
        #pragma unroll
        for (int m = 0; m < MT; ++m)
            #pragma unroll
            for (int t = 0; t < NT; ++t)
                acc[m][t] = __builtin_amdgcn_wmma_f32_16x16x4_f32(
                                false, a0[m], false, b0[t], (short)0,
                                acc[m][t], false, false);
        #pragma unroll
        for (int m = 0; m < MT; ++m) a0[m] = *(const v2f*)(ap[m] + k);
        #pragma unroll
        for (int t = 0; t < NT; ++t) b0[t] = *(const v2f*)(bp[t] + k);

        // consume group 1, refill group 1
        #pragma unroll
        for (int m = 0; m < MT; ++m)
            #pragma unroll
            for (int t = 0; t < NT; ++t)
                acc[m][t] = __builtin_amdgcn_wmma_f32_16x16x4_f32(
                                false, a1[m], false, b1[t], (short)0,
                                acc[m][t], false, false);
        #pragma unroll
        for (int m = 0; m < MT; ++m) a1[m] = *(const v2f*)(ap[m] + k + 4);
        #pragma unroll
        for (int t = 0; t < NT; ++t) b1[t] = *(const v2f*)(bp[t] + k + 4);
    }
    // epilogue: drain both groups
    #pragma unroll
    for (int m = 0; m < MT; ++m)
        #pragma unroll
        for (int t = 0; t < NT; ++t)
            acc[m][t] = __builtin_amdgcn_wmma_f32_16x16x4_f32(
                            false, a0[m], false, b0[t], (short)0,
                            acc[m][t], false, false);
    #pragma unroll
    for (int m = 0; m < MT; ++m)
        #pragma unroll
        for (int t = 0; t < NT; ++t)
            acc[m][t] = __builtin_amdgcn_wmma_f32_16x16x4_f32(
                            false, a1[m], false, b1[t], (short)0,
                            acc[m][t], false, false);

    #pragma unroll
    for (int t = 0; t < NT; ++t) {
        const int col = n0 + (t << 4) + l15;
        if (col < N) {
            const float bb = bias ? bias[col] : 0.f;
            #pragma unroll
            for (int m = 0; m < MT; ++m) {
                #pragma unroll
                for (int r = 0; r < 8; ++r) {
                    const int row = m0 + (m << 4) + r + (half << 3);
                    if (row < M) {
                        long idx = (long)row * N + col;
                        float v = acc[m][t][r] + bb;
                        if (accum) v += Cb[idx];
                        Cb[idx] = v;
                    }
                }
            }
        }
    }
}

// ---------------------------------------------------------------------------
// Elementwise / small kernels
// ---------------------------------------------------------------------------
__global__ void zero_kernel(float* p, int n) {
    int i = blockIdx.x * blockDim.x + threadIdx.x;
    if (i < n) p[i] = 0.f;
}

__global__ void add_kernel(float* out, const float* a, const float* b, int n) {
    int i = blockIdx.x * blockDim.x + threadIdx.x;
    if (i < n) out[i] = a[i] + b[i];
}

__global__ void gelu_kernel(float* x, int n) {
    int i = blockIdx.x * blockDim.x + threadIdx.x;
    if (i < n) {
        float v = x[i];
        x[i] = 0.5f * v * (1.f + erff(v * 0.70710678118654752f));
    }
}

// x_enc[b,t,c] = history[b,t,c,0]  (history stride 5 in last dim)
__global__ void extract_xenc_kernel(const float* __restrict__ hist, float* xenc, int n) {
    int i = blockIdx.x * blockDim.x + threadIdx.x;
    if (i < n) xenc[i] = hist[(long)i * 5];
}

// Moving average (k=25, replicate padding) -> optional mean/residual outputs
__global__ void decomp_kernel(const float* __restrict__ x, float* res,
                              float* mavg, int L, int C, int n)
{
    int i = blockIdx.x * blockDim.x + threadIdx.x;
    if (i >= n) return;
    int c = i % C;
    int l = (i / C) % L;
    int b = i / (C * L);
    const float* xb = x + (long)b * L * C + c;
    float s = 0.f;
    #pragma unroll
    for (int w = -12; w <= 12; ++w) {
        int ll = l + w;
        ll = ll < 0 ? 0 : (ll >= L ? L - 1 : ll);
        s += xb[(long)ll * C];
    }
    float m = s * (1.0f / 25.0f);
    if (mavg) mavg[i] = m;
    if (res)  res[i]  = x[i] - m;
}

// per-(b,c) mean over time axis
__global__ void time_mean_kernel(const float* __restrict__ x, float* cm,
                                 int B, int L, int C)
{
    int i = blockIdx.x * blockDim.x + threadIdx.x;
    if (i >= B * C) return;
    int b = i / C, c = i % C;
    float s = 0.f;
    for (int l = 0; l < L; ++l) s += x[((long)b * L + l) * C + c];
    cm[i] = s / (float)L;
}

__global__ void sub_time_mean_kernel(const float* __restrict__ x,
                                     const float* __restrict__ cm,
                                     float* y, int L, int C, int n)
{
    int i = blockIdx.x * blockDim.x + threadIdx.x;
    if (i >= n) return;
    int c = i % C;
    int b = i / (C * L);
    y[i] = x[i] - cm[b * C + c];
}

// LayerNorm over channels (C==256), one block of 256 per (b,l) row
__global__ void __launch_bounds__(256) layernorm_kernel(
    const float* __restrict__ x, const float* __restrict__ w,
    const float* __restrict__ bb, float* y)
{
    long row = blockIdx.x;
    int c = threadIdx.x;
    float v = x[row * DM + c];
    __shared__ float red[DM];
    __shared__ float mu_s, var_s;
    red[c] = v; __syncthreads();
    for (int o = 128; o > 0; o >>= 1) {
        if (c < o) red[c] += red[c + o];
        __syncthreads();
    }
    if (c == 0) mu_s = red[0] * (1.0f / DM);
    __syncthreads();
    float d = v - mu_s;
    red[c] = d * d; __syncthreads();
    for (int o = 128; o > 0; o >>= 1) {
        if (c < o) red[c] += red[c + o];
        __syncthreads();
    }
    if (c == 0) var_s = red[0] * (1.0f / DM);
    __syncthreads();
    y[row * DM + c] = d * rsqrtf(var_s + 1e-5f) * w[c] + bb[c];
}

// Circular conv (K=3) embedding: out[b,l,d] = sum_c sum_kk src[b,(l+kk-1)%L,c]*W[d,c,kk]
// markmode 0: none; 1: + x_mark_enc @ timeW^T; 2: + x_mark_dec @ timeW^T
__global__ void circconv_kernel(const float* __restrict__ src, int C,
                                const float* __restrict__ Wtok,
                                const float* __restrict__ timeW,
                                const float* __restrict__ hist,
                                const float* __restrict__ fut,
                                int markmode, float* out, int L, int D,
                                int accum, int n)
{
    int i = blockIdx.x * blockDim.x + threadIdx.x;
    if (i >= n) return;
    int d = i % D;
    int l = (i / D) % L;
    int b = i / (D * L);
    int lm = (l == 0) ? L - 1 : l - 1;
    int lp = (l == L - 1) ? 0 : l + 1;
    const float* x0 = src + ((long)b * L + lm) * C;
    const float* x1 = src + ((long)b * L + l) * C;
    const float* x2 = src + ((long)b * L + lp) * C;
    const float* w = Wtok + (long)d * C * 3;
    float s = 0.f;
    for (int c = 0; c < C; ++c)
        s += x0[c] * w[3 * c] + x1[c] * w[3 * c + 1] + x2[c] * w[3 * c + 2];
    if (markmode == 1) {
        const float* mk = hist + (((long)b * LS + l) * NS) * 5 + 1;
        #pragma unroll
        for (int f = 0; f < 4; ++f) s += mk[f] * timeW[(long)d * 4 + f];
    } else if (markmode == 2) {
        const float* mk;
        if (l < LBL) mk = hist + (((long)b * LS + (LS - LBL + l)) * NS) * 5 + 1;
        else         mk = fut  + (((long)b * PRED + (l - LBL)) * NS) * 5 + 1;
        #pragma unroll
        for (int f = 0; f < 4; ++f) s += mk[f] * timeW[(long)d * 4 + f];
    }
    if (accum) out[i] += s; else out[i] = s;
}

// seasonal_init / trend_init assembly
__global__ void build_init_kernel(const float* __restrict__ seas96,
                                  const float* __restrict__ ma96,
                                  const float* __restrict__ xmean,
                                  float* SI, float* TR, int n)
{
    int i = blockIdx.x * blockDim.x + threadIdx.x;
    if (i >= n) return;
    int c = i % NS;
    int l = (i / NS) % LD;
    int b = i / (NS * LD);
    if (l < LBL) {
        long j = ((long)b * LS + (LS - LBL + l)) * NS + c;
        SI[i] = seas96[j];
        TR[i] = ma96[j];
    } else {
        SI[i] = 0.f;
        TR[i] = xmean[b * NS + c];
    }
}

// mean_value[b,tau] = (1/D) * sum_t G[b, t, (t+tau)%L]
__global__ void __launch_bounds__(128) corr_mean_kernel(
    const float* __restrict__ G, float* mv, int L, int D)
{
    int tau = blockIdx.x;
    int b = blockIdx.y;
    __shared__ float red[128];
    const float* Gb = G + (long)b * L * L;
    float s = 0.f;
    for (int t = threadIdx.x; t < L; t += 128) {
        int t2 = t + tau; if (t2 >= L) t2 -= L;
        s += Gb[(long)t * L + t2];
    }
    red[threadIdx.x] = s; __syncthreads();
    for (int o = 64; o > 0; o >>= 1) {
        if (threadIdx.x < o) red[threadIdx.x] += red[threadIdx.x + o];
        __syncthreads();
    }
    if (threadIdx.x == 0) mv[(long)b * L + tau] = red[0] / (float)D;
}

// Iterative top-k (matches jax.lax.top_k ordering: desc value, ties -> lower idx)
// + softmax over the k selected weights. One 256-thread block per batch.
__global__ void __launch_bounds__(256) topk_softmax_kernel(
    const float* __restrict__ meanv, float* wts, int* delays, int L, int K)
{
    int b = blockIdx.x;
    const float* mv = meanv + (long)b * L;
    __shared__ float bestv[256];
    __shared__ int   besti[256];
    __shared__ float selv[TK_STRIDE];
    __shared__ int   seli[TK_STRIDE];
    int tid = threadIdx.x;
    for (int it = 0; it < K; ++it) {
        float bv = -3.4e38f; int bi = -1;
        for (int t = tid; t < L; t += 256) {
            bool used = false;
            for (int j = 0; j < it; ++j) if (seli[j] == t) used = true;
            if (used) continue;
            float v = mv[t];
            if (bi == -1 || v > bv || (v == bv && t < bi)) { bv = v; bi = t; }
        }
        bestv[tid] = bv; besti[tid] = bi;
        __syncthreads();
        for (int o = 128; o > 0; o >>= 1) {
            if (tid < o) {
                float ov = bestv[tid + o]; int oi = besti[tid + o];
                if (oi != -1 && (besti[tid] == -1 || ov > bestv[tid] ||
                    (ov == bestv[tid] && oi < besti[tid]))) {
                    bestv[tid] = ov; besti[tid] = oi;
                }
            }
            __syncthreads();
        }
        if (tid == 0) { selv[it] = bestv[0]; seli[it] = besti[0]; }
        __syncthreads();
    }
    if (tid == 0) {
        float mx = selv[0];  // first pick is the global max
        float s = 0.f;
        for (int j = 0; j < K; ++j) { float e = expf(selv[j] - mx); selv[j] = e; s += e; }
        float inv = 1.f / s;
        for (int j = 0; j < K; ++j) {
            wts[b * TK_STRIDE + j]    = selv[j] * inv;
            delays[b * TK_STRIDE + j] = seli[j];
        }
    }
}

// agg[b,t,d] = sum_i w[b,i] * V[b, (t+delay[b,i])%L, d]
__global__ void agg_gather_kernel(const float* __restrict__ V,
                                  const int* __restrict__ delays,
                                  const float* __restrict__ wts,
                                  float* out, int L, int K, int n)
{
    int i = blockIdx.x * blockDim.x + threadIdx.x;
    if (i >= n) return;
    int d = i % DM;
    int t = (i / DM) % L;
    int b = i / (DM * L);
    float s = 0.f;
    for (int j = 0; j < K; ++j) {
        int tt = t + delays[b * TK_STRIDE + j];
        if (tt >= L) tt -= L;
        s += wts[b * TK_STRIDE + j] * V[((long)b * L + tt) * DM + d];
    }
    out[i] = s;
}

// out[b,t,c] = trend[b,48+t,c] + seasonal_part[b,48+t,c]   (t in [0,720))
__global__ void final_out_kernel(const float* __restrict__ TR,
                                 const float* __restrict__ SP,
                                 float* out, int n)
{
    int i = blockIdx.x * blockDim.x + threadIdx.x;
    if (i >= n) return;
    int c = i % NS;
    int t = (i / NS) % PRED;
    int b = i / (NS * PRED);
    long j = ((long)b * LD + LBL + t) * NS + c;
    out[i] = TR[j] + SP[j];
}

// ---------------------------------------------------------------------------
// Host orchestration
// ---------------------------------------------------------------------------
extern "C" void kernel_launch(void* const* d_in, const int* in_sizes, int n_in,
                              void* d_out, int out_size, void* d_ws, size_t ws_size,
                              hipStream_t stream)
{
    const float* hist        = (const float*)d_in[0];
    const float* fut         = (const float*)d_in[1];
    const float* enc_tok_W   = (const float*)d_in[2];
    const float* dec_tok_W   = (const float*)d_in[3];
    const float* enc_time_W  = (const float*)d_in[4];
    const float* dec_time_W  = (const float*)d_in[5];
    const float* enc_attn_W  = (const float*)d_in[6];
    const float* enc_attn_b  = (const float*)d_in[7];
    const float* enc_ff1_W   = (const float*)d_in[8];
    const float* enc_ff2_W   = (const float*)d_in[9];
    const float* enc_ln_w    = (const float*)d_in[10];
    const float* enc_ln_b    = (const float*)d_in[11];
    const float* dec_self_W  = (const float*)d_in[12];
    const float* dec_self_b  = (const float*)d_in[13];
    const float* dec_cross_W = (const float*)d_in[14];
    const float* dec_cross_b = (const float*)d_in[15];
    const float* dec_ff1_W   = (const float*)d_in[16];
    const float* dec_ff2_W   = (const float*)d_in[17];
    const float* dec_trend_W = (const float*)d_in[18];
    const float* dec_ln_w    = (const float*)d_in[19];
    const float* dec_ln_b    = (const float*)d_in[20];
    const float* proj_W      = (const float*)d_in[21];
    const float* proj_b      = (const float*)d_in[22];
    float* out = (float*)d_out;

    // ---- workspace carve-out (fp32) ----
    float* ws = (float*)d_ws;
    size_t off = 0;
    auto alloc = [&](size_t nf) { float* p = ws + off; off += nf; return p; };
    float* xenc   = alloc((size_t)BN * LS * NS);
    float* seas96 = alloc((size_t)BN * LS * NS);
    float* ma96   = alloc((size_t)BN * LS * NS);
    float* xmean  = alloc((size_t)BN * NS + 16);
    float* SI     = alloc((size_t)BN * LD * NS);
    float* TR     = alloc((size_t)BN * LD * NS);
    float* EO     = alloc((size_t)BN * LS * DM);
    float* Qb     = alloc((size_t)BN * LD * DM);
    float* Kb     = alloc((size_t)BN * LD * DM);
    float* Vb     = alloc((size_t)BN * LD * DM);
    float* AG     = alloc((size_t)BN * LD * DM);
    float* Gm     = alloc((size_t)BN * LD * LD);
    float* meanv  = alloc((size_t)BN * LD);
    float* wts    = alloc((size_t)BN * TK_STRIDE);
    int*   delays = (int*)alloc((size_t)BN * TK_STRIDE);
    float* T1     = alloc((size_t)BN * LD * DM);
    float* T2     = alloc((size_t)BN * LD * DM);
    float* TS     = alloc((size_t)BN * LD * DM);
    float* TT     = alloc((size_t)BN * LD * DM);
    float* MID    = alloc((size_t)BN * LD * DF);
    float* DX     = alloc((size_t)BN * LD * DM);
    float* SP     = alloc((size_t)BN * LD * NS);
    float* cm     = alloc((size_t)BN * DM);

    auto blocks = [](int n) { return dim3((unsigned)((n + 255) / 256)); };

    auto gemm = [&](const float* A, long sA, const float* Bw, long sB,
                    float* C, long sC, const float* bias,
                    int M, int N, int K, int Z, int accum) {
        dim3 g((unsigned)((N + 63) / 64), (unsigned)((M + 31) / 32), (unsigned)Z);
        gemm_xwt_wmma<<<g, dim3(32), 0, stream>>>(A, sA, Bw, sB, C, sC, bias,
                                                  M, N, K, accum);
    };

    // AutoCorrelation layer: out = ACL(xq(Lq), xkv(S)); weights W[4][DM][DM], b[4][DM]
    auto run_acl = [&](const float* xq, int Lq, const float* xkv, int S,
                       const float* W, const float* bv, float* outb, int topk) {
        const float* W0 = W;
        const float* W1 = W + (size_t)DM * DM;
        const float* W2 = W + (size_t)2 * DM * DM;
        const float* W3 = W + (size_t)3 * DM * DM;
        const float* b0 = bv;
        const float* b1 = bv + DM;
        const float* b2 = bv + 2 * DM;
        const float* b3 = bv + 3 * DM;
        // Q projection
        gemm(xq, 0, W0, 0, Qb, 0, b0, BN * Lq, DM, DM, 1, 0);
        if (S == Lq) {
            gemm(xkv, 0, W1, 0, Kb, 0, b1, BN * S, DM, DM, 1, 0);
            gemm(xkv, 0, W2, 0, Vb, 0, b2, BN * S, DM, DM, 1, 0);
        } else { // zero-pad K,V up to Lq per batch (reference pads with zeros)
            int nz = BN * Lq * DM;
            zero_kernel<<<blocks(nz), 256, 0, stream>>>(Kb, nz);
            zero_kernel<<<blocks(nz), 256, 0, stream>>>(Vb, nz);
            gemm(xkv, (long)S * DM, W1, 0, Kb, (long)Lq * DM, b1, S, DM, DM, BN, 0);
            gemm(xkv, (long)S * DM, W2, 0, Vb, (long)Lq * DM, b2, S, DM, DM, BN, 0);
        }
        // Gram matrix G[b] = Q[b] @ K[b]^T  (replaces FFT autocorrelation)
        gemm(Qb, (long)Lq * DM, Kb, (long)Lq * DM, Gm, (long)Lq * Lq,
             nullptr, Lq, Lq, DM, BN, 0);
        corr_mean_kernel<<<dim3((unsigned)Lq, BN), 128, 0, stream>>>(Gm, meanv, Lq, DM);
        topk_softmax_kernel<<<BN, 256, 0, stream>>>(meanv, wts, delays, Lq, topk);
        int na = BN * Lq * DM;
        agg_gather_kernel<<<blocks(na), 256, 0, stream>>>(Vb, delays, wts, AG, Lq, topk, na);
        gemm(AG, 0, W3, 0, outb, 0, b3, BN * Lq, DM, DM, 1, 0);
    };

    auto run_ffn = [&](const float* x, int L, const float* Wf1, const float* Wf2,
                       float* outb) {
        gemm(x, 0, Wf1, 0, MID, 0, nullptr, BN * L, DF, DM, 1, 0);
        int n = BN * L * DF;
        gelu_kernel<<<blocks(n), 256, 0, stream>>>(MID, n);
        gemm(MID, 0, Wf2, 0, outb, 0, nullptr, BN * L, DM, DF, 1, 0);
    };

    auto run_ln = [&](float* x, int L, const float* w, const float* bb, float* y) {
        layernorm_kernel<<<BN * L, 256, 0, stream>>>(x, w, bb, y);
        time_mean_kernel<<<blocks(BN * DM), 256, 0, stream>>>(y, cm, BN, L, DM);
        int n = BN * L * DM;
        sub_time_mean_kernel<<<blocks(n), 256, 0, stream>>>(y, cm, y, L, DM, n);
    };

    // =========================== setup ===========================
    int nx = BN * LS * NS;
    extract_xenc_kernel<<<blocks(nx), 256, 0, stream>>>(hist, xenc, nx);
    decomp_kernel<<<blocks(nx), 256, 0, stream>>>(xenc, seas96, ma96, LS, NS, nx);
    time_mean_kernel<<<blocks(BN * NS), 256, 0, stream>>>(xenc, xmean, BN, LS, NS);
    int ni = BN * LD * NS;
    build_init_kernel<<<blocks(ni), 256, 0, stream>>>(seas96, ma96, xmean, SI, TR, ni);

    // =========================== encoder =========================
    int nE = BN * LS * DM;
    circconv_kernel<<<blocks(nE), 256, 0, stream>>>(xenc, NS, enc_tok_W, enc_time_W,
                                                    hist, nullptr, 1, EO, LS, DM, 0, nE);
    for (int l = 0; l < 2; ++l) {
        const float* Wl = enc_attn_W + (size_t)l * 4 * DM * DM;
        const float* bl = enc_attn_b + (size_t)l * 4 * DM;
        run_acl(EO, LS, EO, LS, Wl, bl, T1, TOPK_ENC);
        add_kernel<<<blocks(nE), 256, 0, stream>>>(T2, EO, T1, nE);
        decomp_kernel<<<blocks(nE), 256, 0, stream>>>(T2, T1, nullptr, LS, DM, nE);
        run_ffn(T1, LS, enc_ff1_W + (size_t)l * DF * DM,
                enc_ff2_W + (size_t)l * DM * DF, T2);
        add_kernel<<<blocks(nE), 256, 0, stream>>>(T2, T1, T2, nE);
        decomp_kernel<<<blocks(nE), 256, 0, stream>>>(T2, EO, nullptr, LS, DM, nE);
    }
    run_ln(EO, LS, enc_ln_w, enc_ln_b, EO);

    // =========================== decoder =========================
    int nD = BN * LD * DM;
    circconv_kernel<<<blocks(nD), 256, 0, stream>>>(SI, NS, dec_tok_W, dec_time_W,
                                                    hist, fut, 2, DX, LD, DM, 0, nD);
    // D_LAYERS == 1
    {
        // self autocorrelation
        run_acl(DX, LD, DX, LD, dec_self_W, dec_self_b, T1, TOPK_DEC);
        add_kernel<<<blocks(nD), 256, 0, stream>>>(T2, DX, T1, nD);
        decomp_kernel<<<blocks(nD), 256, 0, stream>>>(T2, DX, TS, LD, DM, nD);
        // cross autocorrelation vs encoder output (S=96, zero-padded to 768)
        run_acl(DX, LD, EO, LS, dec_cross_W, dec_cross_b, T1, TOPK_DEC);
        add_kernel<<<blocks(nD), 256, 0, stream>>>(T2, DX, T1, nD);
        decomp_kernel<<<blocks(nD), 256, 0, stream>>>(T2, DX, TT, LD, DM, nD);
        add_kernel<<<blocks(nD), 256, 0, stream>>>(TS, TS, TT, nD);
        // FFN
        run_ffn(DX, LD, dec_ff1_W, dec_ff2_W, T1);
        add_kernel<<<blocks(nD), 256, 0, stream>>>(T2, DX, T1, nD);
        decomp_kernel<<<blocks(nD), 256, 0, stream>>>(T2, DX, TT, LD, DM, nD);
        add_kernel<<<blocks(nD), 256, 0, stream>>>(TS, TS, TT, nD);
        // trend accumulation: trend += circ_conv(t1+t2+t3, dec_trend_W[0])
        int nT = BN * LD * NS;
        circconv_kernel<<<blocks(nT), 256, 0, stream>>>(TS, DM, dec_trend_W, nullptr,
                                                        nullptr, nullptr, 0, TR, LD, NS,
                                                        1, nT);
    }
    run_ln(DX, LD, dec_ln_w, dec_ln_b, DX);
    // seasonal projection: (B*768 x 256) @ (321 x 256)^T + b   (N=321: guarded tiles)
    gemm(DX, 0, proj_W, 0, SP, 0, proj_b, BN * LD, NS, DM, 1, 0);

    int no = BN * PRED * NS;
    final_out_kernel<<<blocks(no), 256, 0, stream>>>(TR, SP, out, no);
}